// MLPAttention_52381421142032
// MI455X (gfx1250) — compile-verified
//
#include <hip/hip_runtime.h>
#include <hip/hip_bf16.h>

typedef __attribute__((ext_vector_type(2))) float v2f;
typedef __attribute__((ext_vector_type(4))) float v4f;
typedef __attribute__((ext_vector_type(8))) float v8f;
typedef __attribute__((ext_vector_type(4))) unsigned int v4u;
typedef __attribute__((ext_vector_type(4))) int v4i;
typedef __attribute__((ext_vector_type(8))) int v8i;

#define B_ 8
#define L_ 512
#define D_ 128
#define LDS_STRIDE 132   // 132 % 64 == 4 -> conflict-free strided column reads; rows 16B aligned

__device__ __forceinline__ float tanh_fast(float x) {
  // tanh(x) = 1 - 2/(exp2(2*log2(e)*x) + 1) ; saturates correctly at +-1
  float e = __builtin_amdgcn_exp2f(x * 2.885390081777927f);
  return 1.0f - 2.0f * __builtin_amdgcn_rcpf(e + 1.0f);
}

// Stage a 16x128 f32 tile (row-major, row stride 128) from global into LDS with
// row stride LDS_STRIDE floats, using gfx1250 async global->LDS b128 copies.
// All 256 threads participate; caller must s_wait_asynccnt + barrier.
__device__ __forceinline__ void async_stage_tile(uint64_t gbase, uint32_t ldsbase, int tid) {
#pragma unroll
  for (int it = 0; it < 2; ++it) {
    const int f   = tid + it * 256;          // float4 index: 16 rows * 32 float4
    const int row = f >> 5;
    const int c4  = f & 31;
    const uint32_t voff = (uint32_t)(row * (D_ * 4) + c4 * 16);           // global byte offset
    const uint32_t lds  = ldsbase + (uint32_t)(row * (LDS_STRIDE * 4) + c4 * 16); // LDS byte addr
    asm volatile("global_load_async_to_lds_b128 %0, %1, %2"
                 :: "v"(lds), "v"(voff), "s"(gbase) : "memory");
  }
}

__device__ __forceinline__ void wait_async_lds() {
  asm volatile("s_wait_asynccnt 0x0" ::: "memory");
}

// Kernel 1: qbuf = inp @ Wq^T + bq ; kbuf = inp @ Wk^T
// grid = (B*L)/16 blocks, 256 threads (8 waves); wave w -> output cols [16w,16w+16)
__global__ __launch_bounds__(256) void qk_proj_kernel(
    const float* __restrict__ inp, const float* __restrict__ Wq,
    const float* __restrict__ bq, const float* __restrict__ Wk,
    float* __restrict__ qbuf, float* __restrict__ kbuf) {
  const int tid  = threadIdx.x;
  const int lane = tid & 31;
  const int wave = tid >> 5;
  const int m    = lane & 15;   // M (A) / N (B,C,D) index
  const int kh   = lane >> 4;   // K half-select
  const int rowBase = blockIdx.x * 16;  // over B*L
  const int colBase = wave * 16;        // over D
  v8f accq = {};
  v8f acck = {};
  for (int kk = 0; kk < D_; kk += 4) {
    const int kidx = kk + 2 * kh;
    v2f a, bqv, bkv;
    a.x = inp[(rowBase + m) * D_ + kidx];
    a.y = inp[(rowBase + m) * D_ + kidx + 1];
    bqv.x = Wq[(colBase + m) * D_ + kidx];
    bqv.y = Wq[(colBase + m) * D_ + kidx + 1];
    bkv.x = Wk[(colBase + m) * D_ + kidx];
    bkv.y = Wk[(colBase + m) * D_ + kidx + 1];
    accq = __builtin_amdgcn_wmma_f32_16x16x4_f32(false, a, false, bqv, (short)0, accq, false, false);
    acck = __builtin_amdgcn_wmma_f32_16x16x4_f32(false, a, false, bkv, (short)0, acck, false, false);
  }
  const float bias = bq[colBase + m];
#pragma unroll
  for (int r = 0; r < 8; ++r) {
    const int row = rowBase + r + 8 * kh;   // C/D layout: M = r + 8*(lane/16)
    qbuf[row * D_ + colBase + m] = accq[r] + bias;
    kbuf[row * D_ + colBase + m] = acck[r];
  }
}

// Kernel 2: fused score -> softmax -> out for one (b, 16-row i-tile)
// grid = B * (L/16) blocks, 256 threads (8 waves)
__global__ __launch_bounds__(256) void attn_kernel(
    const float* __restrict__ inp, const float* __restrict__ qbuf,
    const float* __restrict__ kbuf, const float* __restrict__ vvec,
    float* __restrict__ out, float* __restrict__ attn) {
  __shared__ __align__(16) float sK[16][LDS_STRIDE];
  __shared__ __align__(16) float sQ[16][LDS_STRIDE];   // q staging, later inp staging
  __shared__ __align__(16) float sV[D_];
  __shared__ __align__(16) float sS[16][516];          // 516 % 64 == 4 -> conflict-free WMMA-A reads

  const int tid = threadIdx.x;
  const int tx  = tid & 15;       // j offset within 16-wide stripe
  const int ty  = tid >> 4;       // i row within tile (half-wave aligned)
  const int b   = blockIdx.x >> 5;
  const int i0  = (blockIdx.x & 31) * 16;

  if (tid < D_) sV[tid] = vvec[tid];

  // ---- k-tile via Tensor Data Mover: 2D D# (tile 128x16 f32), pad_interval=128 DWORDs,
  // pad_amount=4 DWORDs -> TDM writes directly into the 132-float padded LDS layout.
  if (tid < 32) {  // wave 0 issues one TDM op for the block (EXEC-independent)
    const uint64_t ga  = (uint64_t)(uintptr_t)(kbuf + (size_t)(b * L_ + i0) * D_);
    const uint32_t lds = (uint32_t)(uintptr_t)&sK[0][0];
    v4u g0 = { 1u,                                  // count=1 (valid user descriptor)
               lds,                                 // lds_addr
               (unsigned int)ga,                    // global_addr[31:0]
               (unsigned int)(((ga >> 32) & 0x1FFFFFFu) | (2u << 30)) }; // addr[56:32] | type=2
    v8i g1 = { (int)((2u << 16) | (1u << 20) | (6u << 22) | (3u << 25)),
               //  data_size=4B | pad_enable | pad_interval=128dw | pad_amount=4dw
               (int)(128u << 16),                   // tensor_dim0 = 128 (lo16)
               (int)(512u << 16),                   // tensor_dim1 = 512 (lo16)
               (int)(128u << 16),                   // tile_dim0 = 128
               16,                                  // tile_dim1 = 16, tile_dim2 = 0
               128,                                 // tensor_dim0_stride = 128
               0, 0 };
    v4i gz4 = { 0, 0, 0, 0 };
    v8i gz8 = { 0, 0, 0, 0, 0, 0, 0, 0 };
    __builtin_amdgcn_tensor_load_to_lds(g0, g1, gz4, gz4, gz8, 0);
    __builtin_amdgcn_s_wait_tensorcnt(0);
  }
  __syncthreads();

  // ---- Phase 1: score[i0+ty][j] = sum_d v[d]*tanh(q[j][d] + k[i0+ty][d])
  const uint32_t sQ_lds = (uint32_t)(uintptr_t)&sQ[0][0];
  for (int j0 = 0; j0 < L_; j0 += 16) {
    async_stage_tile((uint64_t)(uintptr_t)(qbuf + (size_t)(b * L_ + j0) * D_), sQ_lds, tid);
    wait_async_lds();
    __syncthreads();
    float acc = 0.f;
    const v4f* qr = (const v4f*)sQ[tx];
    const v4f* kr = (const v4f*)sK[ty];
    const v4f* vr = (const v4f*)sV;
#pragma unroll 4
    for (int d4 = 0; d4 < D_ / 4; ++d4) {
      v4f qv = qr[d4], kv = kr[d4], vv = vr[d4];
      acc += vv.x * tanh_fast(qv.x + kv.x);
      acc += vv.y * tanh_fast(qv.y + kv.y);
      acc += vv.z * tanh_fast(qv.z + kv.z);
      acc += vv.w * tanh_fast(qv.w + kv.w);
    }
    sS[ty][j0 + tx] = acc;
    __syncthreads();
  }

  // ---- Phase 2: softmax over j (row ty handled by one half-wave of 16 lanes)
  float mx = -3.0e38f;
  for (int c = tx; c < L_; c += 16) mx = fmaxf(mx, sS[ty][c]);
#pragma unroll
  for (int off = 1; off < 16; off <<= 1) mx = fmaxf(mx, __shfl_xor(mx, off, 32));
  float sum = 0.f;
  for (int c = tx; c < L_; c += 16) {
    float e = __builtin_amdgcn_exp2f((sS[ty][c] - mx) * 1.4426950408889634f);
    sS[ty][c] = e;
    sum += e;
  }
#pragma unroll
  for (int off = 1; off < 16; off <<= 1) sum += __shfl_xor(sum, off, 32);
  const float inv = __builtin_amdgcn_rcpf(sum);
  for (int c = tx; c < L_; c += 16) {
    float a = sS[ty][c] * inv;
    sS[ty][c] = a;
    attn[(size_t)(b * L_ + i0 + ty) * L_ + c] = a;
  }
  __syncthreads();

  // ---- Phase 3: out[i0:i0+16, :] = attn_tile[16x512] @ inp[b][512x128]  (f32 WMMA)
  const int lane = tid & 31;
  const int wave = tid >> 5;
  const int n    = lane & 15;
  const int kh   = lane >> 4;
  const int colBase = wave * 16;
  v8f acc = {};
  for (int chunk = 0; chunk < L_; chunk += 16) {
    __syncthreads();  // previous staging consumed
    async_stage_tile((uint64_t)(uintptr_t)(inp + (size_t)(b * L_ + chunk) * D_), sQ_lds, tid);
    wait_async_lds();
    __syncthreads();
#pragma unroll
    for (int kk = 0; kk < 16; kk += 4) {
      v2f a, bb;
      a.x  = sS[n][chunk + kk + 2 * kh];          // A[m=n][k]
      a.y  = sS[n][chunk + kk + 2 * kh + 1];
      bb.x = sQ[kk + 2 * kh][colBase + n];        // B[k][n]
      bb.y = sQ[kk + 2 * kh + 1][colBase + n];
      acc = __builtin_amdgcn_wmma_f32_16x16x4_f32(false, a, false, bb, (short)0, acc, false, false);
    }
  }
#pragma unroll
  for (int r = 0; r < 8; ++r)
    out[(size_t)(b * L_ + i0 + r + 8 * kh) * D_ + colBase + n] = acc[r];
}

extern "C" void kernel_launch(void* const* d_in, const int* in_sizes, int n_in,
                              void* d_out, int out_size, void* d_ws, size_t ws_size,
                              hipStream_t stream) {
  const float* inp = (const float*)d_in[0];
  const float* Wq  = (const float*)d_in[1];
  const float* bq  = (const float*)d_in[2];
  const float* Wk  = (const float*)d_in[3];
  const float* v   = (const float*)d_in[4];

  float* out  = (float*)d_out;                        // [B,L,D]
  float* attn = (float*)d_out + (size_t)B_ * L_ * D_; // [B,L,L]

  float* qbuf = (float*)d_ws;                         // [B*L, D]
  float* kbuf = qbuf + (size_t)B_ * L_ * D_;          // [B*L, D]

  qk_proj_kernel<<<(B_ * L_) / 16, 256, 0, stream>>>(inp, Wq, bq, Wk, qbuf, kbuf);
  attn_kernel<<<B_ * (L_ / 16), 256, 0, stream>>>(inp, qbuf, kbuf, v, out, attn);
}